// AutoCorrelationLite_29781303231199
// MI455X (gfx1250) — compile-verified
//
#include <hip/hip_runtime.h>
#include <hip/hip_bf16.h>

typedef __attribute__((ext_vector_type(2))) float v2f;
typedef __attribute__((ext_vector_type(8))) float v8f;

#define L_DIM  4096
#define D_DIM  512
#define B_DIM  32
#define NUM_LAGS 32
#define K_TOP 5

// ---------------------------------------------------------------------------
// Pass 1: x_mean[b,l] = mean over D (512) using V_WMMA_F32_16X16X4_F32.
// One wave handles 16 rows. Lane (m = lane&15, half = lane>>4) streams the
// contiguous half-row x[row, half*256 .. half*256+255] with float4 loads and
// pre-reduces into the 16x4 f32 A-matrix layout (lane half supplies K=2h,2h+1).
// One WMMA vs an all-(1/512) B matrix then yields all 16 row means.
// ---------------------------------------------------------------------------
__global__ void __launch_bounds__(128)
mean_wmma_kernel(const float* __restrict__ x, float* __restrict__ mean) {
    const int lane = threadIdx.x & 31;
    const int wave = threadIdx.x >> 5;
    const int w    = blockIdx.x * 4 + wave;      // global wave id
    const int row16 = w * 16;                    // first of 16 rows
    const int m    = lane & 15;
    const int half = lane >> 4;

    const float4* __restrict__ p =
        (const float4*)(x + (size_t)(row16 + m) * D_DIM + half * 256);

    float sx = 0.f, sy = 0.f, sz = 0.f, sw = 0.f;
#pragma unroll 8
    for (int c = 0; c < 64; ++c) {
        float4 v = p[c];
        sx += v.x; sy += v.y; sz += v.z; sw += v.w;
    }

    v2f a; a.x = sx + sy; a.y = sz + sw;          // A[m][2h], A[m][2h+1]
    v2f b; b.x = 1.0f / 512.0f; b.y = 1.0f / 512.0f;  // B = ones/512
    v8f c = {};
    c = __builtin_amdgcn_wmma_f32_16x16x4_f32(
            /*neg_a=*/false, a, /*neg_b=*/false, b,
            /*c_mod=*/(short)0, c, /*reuse_a=*/false, /*reuse_b=*/false);

    // D layout: VGPR v, lanes 0-15 -> M=v; lanes 16-31 -> M=v+8 (replicated N).
    if (lane == 0) {
#pragma unroll
        for (int v = 0; v < 8; ++v) mean[row16 + v] = c[v];
    } else if (lane == 16) {
#pragma unroll
        for (int v = 0; v < 8; ++v) mean[row16 + 8 + v] = c[v];
    }
}

// ---------------------------------------------------------------------------
// Pass 2: one wave per batch. Lane j computes lag_j and its autocorrelation
// score; lane 0 does top-5 (strict '>' scan = first-index tie-break like
// lax.top_k) and writes weights + selected lags.
// ---------------------------------------------------------------------------
__global__ void __launch_bounds__(32)
score_topk_kernel(const float* __restrict__ mean,
                  float* __restrict__ weights, int* __restrict__ sel_lags) {
    const int b = blockIdx.x;
    const int j = threadIdx.x;                    // 0..31

    // linspace(1, 168, 32) truncated toward zero
    const int lag = (int)(1.0 + (167.0 * (double)j) / 31.0);

    const float* __restrict__ mb = mean + b * L_DIM;
    float s = 0.f;
    const int n = L_DIM - lag;
    for (int t = 0; t < n; ++t) s += mb[t] * mb[t + lag];
    s /= (float)n;

    __shared__ float sc[NUM_LAGS];
    __shared__ int   lg[NUM_LAGS];
    sc[j] = s; lg[j] = lag;
    __syncthreads();

    if (j == 0) {
        float vals[K_TOP]; int lsel[K_TOP];
        float sum = 0.f;
#pragma unroll
        for (int k = 0; k < K_TOP; ++k) {
            int best = 0; float bv = -3.402823466e38f;
            for (int i = 0; i < NUM_LAGS; ++i)
                if (sc[i] > bv) { bv = sc[i]; best = i; }
            vals[k] = bv; lsel[k] = lg[best];
            sc[best] = -3.402823466e38f;
            sum += bv;
        }
        const float den = sum + 1e-6f;
#pragma unroll
        for (int k = 0; k < K_TOP; ++k) {
            weights[b * K_TOP + k]  = vals[k] / den;
            sel_lags[b * K_TOP + k] = lsel[k];
        }
    }
}

// ---------------------------------------------------------------------------
// Pass 3: out[b,l,:] = sum_j w[b,j] * x[b, (l - lag_j) mod L, :]
// 256 threads = 2 rows per block, float4 (b128) loads/stores. Blocks are
// batch-major so the 5 shifted reads of the 8 MB batch slice stay in L2.
// ---------------------------------------------------------------------------
__global__ void __launch_bounds__(256)
combine_kernel(const float* __restrict__ x,
               const float* __restrict__ weights,
               const int* __restrict__ sel_lags,
               float* __restrict__ out) {
    const int tid = threadIdx.x;
    const int c   = tid & 127;                    // float4 column 0..127
    const int r   = tid >> 7;                     // row-in-block 0..1
    const int b   = blockIdx.x >> 11;             // / (L/2)
    const int lp  = blockIdx.x & 2047;
    const int l   = lp * 2 + r;

    const float4* __restrict__ xb =
        (const float4*)(x + (size_t)b * L_DIM * D_DIM);

    float ax = 0.f, ay = 0.f, az = 0.f, aw = 0.f;
#pragma unroll
    for (int k = 0; k < K_TOP; ++k) {
        const int   lag = sel_lags[b * K_TOP + k];
        const float wk  = weights[b * K_TOP + k];
        int src = l - lag; if (src < 0) src += L_DIM;
        float4 v = xb[(size_t)src * 128 + c];
        ax += wk * v.x; ay += wk * v.y; az += wk * v.z; aw += wk * v.w;
    }

    float4 o; o.x = ax; o.y = ay; o.z = az; o.w = aw;
    ((float4*)out)[((size_t)b * L_DIM + l) * 128 + c] = o;
}

// ---------------------------------------------------------------------------
extern "C" void kernel_launch(void* const* d_in, const int* in_sizes, int n_in,
                              void* d_out, int out_size, void* d_ws, size_t ws_size,
                              hipStream_t stream) {
    const float* x = (const float*)d_in[0];
    float* out = (float*)d_out;

    // workspace layout
    float* mean    = (float*)d_ws;                              // 512 KB
    float* weights = (float*)((char*)d_ws + 512 * 1024);        // 640 B
    int*   lags    = (int*)  ((char*)d_ws + 512 * 1024 + 1024); // 640 B

    // Pass 1: B*L/16 = 8192 waves -> 2048 blocks of 128 threads
    mean_wmma_kernel<<<2048, 128, 0, stream>>>(x, mean);
    // Pass 2: one wave per batch
    score_topk_kernel<<<B_DIM, 32, 0, stream>>>(mean, weights, lags);
    // Pass 3: B * L/2 = 65536 blocks
    combine_kernel<<<B_DIM * (L_DIM / 2), 256, 0, stream>>>(x, weights, lags, out);
}